// FogGenerator_20538533610277
// MI455X (gfx1250) — compile-verified
//
#include <hip/hip_runtime.h>
#include <cstdint>

// ---------------------------------------------------------------------------
// Fog generator (guided filter dehaze-style compositing) for MI455X / gfx1250.
// Horizontal 81-tap box sums are computed with V_WMMA_F32_16X16X4_F32 as a
// banded-ones matmul (full f32 precision). Image-boundary handling is done
// with zeroed guard columns in LDS, and the banded-ones B fragments are
// translation-invariant, precomputed once per wave and kept register-resident
// (k-tile loop fully unrolled so no scratch spill) -> the steady-state inner
// loop is ds_load + v_mul + v_wmma only. Vertical passes are per-column
// running sums (subtract re-reads stay L2-hot; working set ~13MB << 192MB L2).
// Top-k atmospheric light via deterministic 4-round radix select on float
// bits (integer atomics) + fixed-order float reductions (graph-replay safe).
// ---------------------------------------------------------------------------

typedef float v2f __attribute__((ext_vector_type(2)));
typedef float v8f __attribute__((ext_vector_type(8)));

namespace {
constexpr int   BN        = 8;
constexpr int   HN        = 768;
constexpr int   WN        = 1280;
constexpr int   HWN       = HN * WN;          // 983040
constexpr int   RAD       = 40;               // 81x81 window
constexpr float INV_N     = 1.0f / 6561.0f;   // (2r+1)^2
constexpr float EPS_F     = 1e-3f;
constexpr float BETA_F    = 2.996f / 300.0f;
constexpr int   KTOP      = 983;              // max(10, int(0.001*H*W))
constexpr int   GUARD     = 40;               // zero guard columns each side
constexpr int   LDSW      = GUARD + WN + GUARD;   // 1360
constexpr int   LDS_STRIDE = LDSW + 1;        // 1361 (== 17 mod 64: no bank conflicts)
constexpr int   NCHUNK    = 4;                // vertical-pass row chunks
constexpr int   CHUNK     = HN / NCHUNK;      // 192
constexpr int   PBLK      = 240;              // partial-sum blocks per image
constexpr int   KT_PER_TILE = 24;             // (40+16+40)/4 banded K-tiles
}

// D = A(16x4,f32) * B(4x16,f32) + C(16x16,f32), one wave32.
__device__ __forceinline__ v8f wmma_acc(v8f c, v2f a, v2f b) {
  // args: (neg_a, A, neg_b, B, c_mod, C, reuse_a, reuse_b)
  return __builtin_amdgcn_wmma_f32_16x16x4_f32(false, a, false, b,
                                               (short)0, c, false, false);
}

// Zero the guard columns of two LDS fields (cooperative, pre-barrier).
__device__ __forceinline__ void zero_guards(float* f0, float* f1) {
  for (int idx = threadIdx.x; idx < 16 * 2 * GUARD; idx += 256) {
    const int row = idx / (2 * GUARD);
    const int gi  = idx - row * (2 * GUARD);
    const int col = (gi < GUARD) ? gi : (GUARD + WN + (gi - GUARD));
    f0[row * LDS_STRIDE + col] = 0.0f;
    f1[row * LDS_STRIDE + col] = 0.0f;
  }
}

// Precompute the 24 banded-ones B fragments for this lane. With guard
// columns handling image bounds, the indicator depends only on
// u = 4*i + khalf - ncol:  bb.x = [0 <= u <= 80], bb.y = [-1 <= u <= 79].
__device__ __forceinline__ void make_band(v2f* bbA, int khalf, int ncol) {
#pragma unroll
  for (int i = 0; i < KT_PER_TILE; ++i) {
    const int u = 4 * i + khalf - ncol;
    bbA[i].x = (u >= 0 && u <= 80) ? 1.0f : 0.0f;
    bbA[i].y = (u >= -1 && u <= 79) ? 1.0f : 0.0f;
  }
}

// ---------------------------------------------------------------------------
// Kernel A: compute gray & t_raw for a 16-row band into LDS (with zero
// guards), then horizontal 81-tap box sums of {I, p, I*p, I*I} via WMMA.
// Grid: (H/16, B), block 256 (8 waves, 10 col-tiles each).
// ---------------------------------------------------------------------------
__global__ void prep_hbox(const float* __restrict__ img,
                          const float* __restrict__ depth,
                          float* __restrict__ fI,  float* __restrict__ fP,
                          float* __restrict__ fIP, float* __restrict__ fII) {
  extern __shared__ float lds[];
  float* gL = lds;                       // [16][LDS_STRIDE]
  float* pL = lds + 16 * LDS_STRIDE;

  const int band = blockIdx.x;
  const int b    = blockIdx.y;
  const int r0   = band * 16;

  zero_guards(gL, pL);
  // Cooperative load: 16 rows of gray and t_raw into LDS (offset by GUARD).
  for (int idx = threadIdx.x; idx < 16 * WN; idx += 256) {
    const int rr = idx / WN;
    const int cc = idx - rr * WN;
    const int y  = r0 + rr;
    const size_t base = ((size_t)(b * 3) * HN + y) * WN + cc;
    const float R  = img[base];
    const float G  = img[base + (size_t)HN * WN];
    const float Bc = img[base + 2 * (size_t)HN * WN];
    const float g = 0.299f * R + 0.587f * G + 0.114f * Bc;
    const float d = depth[((size_t)b * HN + y) * WN + cc];
    float p = expf(-BETA_F * d);
    p = fminf(fmaxf(p, 0.0f), 1.0f);
    gL[rr * LDS_STRIDE + GUARD + cc] = g;
    pL[rr * LDS_STRIDE + GUARD + cc] = p;
  }
  __syncthreads();

  const int lane  = threadIdx.x & 31;
  const int wave  = threadIdx.x >> 5;
  const int Mrow  = lane & 15;            // A: M = lane%16 (both halves)
  const int khalf = (lane >> 4) << 1;     // A/B VGPR0 holds K=0(lo)/K=2(hi)
  const int ncol  = lane & 15;            // B/D: N = lane%16
  const int rowOff = (lane >> 4) << 3;    // D: lanes16-31 -> M+8

  v2f bbA[KT_PER_TILE];
  make_band(bbA, khalf, ncol);
  const int arow = Mrow * LDS_STRIDE + GUARD;  // lane's A-row base in LDS

  for (int t = 0; t < 10; ++t) {
    const int c0 = (wave * 10 + t) * 16;
    v8f accI  = {0,0,0,0,0,0,0,0};
    v8f accP  = {0,0,0,0,0,0,0,0};
    v8f accIP = {0,0,0,0,0,0,0,0};
    v8f accII = {0,0,0,0,0,0,0,0};

    const int kbase = arow + c0 - RAD + khalf;  // i=0 slot for this lane
#pragma unroll
    for (int i = 0; i < KT_PER_TILE; ++i) {     // full unroll: bbA stays in VGPRs
      const int k = kbase + 4 * i;
      const float g0 = gL[k],     g1 = gL[k + 1];
      const float p0 = pL[k],     p1 = pL[k + 1];
      v2f aI  = {g0, g1};
      v2f aP  = {p0, p1};
      v2f aIP = {g0 * p0, g1 * p1};
      v2f aII = {g0 * g0, g1 * g1};
      accI  = wmma_acc(accI,  aI,  bbA[i]);
      accP  = wmma_acc(accP,  aP,  bbA[i]);
      accIP = wmma_acc(accIP, aIP, bbA[i]);
      accII = wmma_acc(accII, aII, bbA[i]);
    }

#pragma unroll
    for (int j = 0; j < 8; ++j) {
      const int row = r0 + j + rowOff;
      const size_t o = ((size_t)b * HN + row) * WN + (c0 + ncol);
      fI[o]  = accI[j];
      fP[o]  = accP[j];
      fIP[o] = accIP[j];
      fII[o] = accII[j];
    }
  }
}

// ---------------------------------------------------------------------------
// Kernel B: vertical 81-tap box of the 4 fields (running sums down columns),
// then a = cov/(var+eps), b = mean_p - a*mean_I.
// One thread per (image, column, row-chunk). Subtract re-reads stay L2-hot.
// ---------------------------------------------------------------------------
__global__ void vbox_ab(const float* __restrict__ fI,  const float* __restrict__ fP,
                        const float* __restrict__ fIP, const float* __restrict__ fII,
                        float* __restrict__ aF, float* __restrict__ bF) {
  const int gid = blockIdx.x * blockDim.x + threadIdx.x;
  if (gid >= BN * WN * NCHUNK) return;
  const int x     = gid % WN;
  const int rest  = gid / WN;
  const int b     = rest % BN;
  const int chunk = rest / BN;
  const int y0    = chunk * CHUNK;
  const size_t base = (size_t)b * HN * WN + x;

  float sI = 0.f, sP = 0.f, sIP = 0.f, sII = 0.f;
  for (int yy = y0 - RAD; yy < y0 + RAD; ++yy) {
    if (yy >= 0 && yy < HN) {
      const size_t o = base + (size_t)yy * WN;
      sI += fI[o]; sP += fP[o]; sIP += fIP[o]; sII += fII[o];
    }
  }
  for (int y = y0; y < y0 + CHUNK; ++y) {
    const int ya = y + RAD;
    if (ya < HN) {
      const size_t o = base + (size_t)ya * WN;
      sI += fI[o]; sP += fP[o]; sIP += fIP[o]; sII += fII[o];
      if (ya + 16 < HN) {  // global_prefetch_b8 ahead of the streaming front
        __builtin_prefetch(fI + base + (size_t)(ya + 16) * WN, 0, 1);
        __builtin_prefetch(fII + base + (size_t)(ya + 16) * WN, 0, 1);
      }
    }
    const float mI  = sI * INV_N,  mP  = sP * INV_N;
    const float mIP = sIP * INV_N, mII = sII * INV_N;
    const float cov = mIP - mI * mP;
    const float var = mII - mI * mI;
    const float a   = cov / (var + EPS_F);
    const float bb  = mP - a * mI;
    const size_t oo = base + (size_t)y * WN;
    aF[oo] = a;
    bF[oo] = bb;
    const int ys = y - RAD;
    if (ys >= 0) {
      const size_t o = base + (size_t)ys * WN;
      sI -= fI[o]; sP -= fP[o]; sIP -= fIP[o]; sII -= fII[o];
    }
  }
}

// ---------------------------------------------------------------------------
// Kernel C: horizontal box of a and b via WMMA, in place (rows fully staged
// in LDS before stores; h-pass is row-local so no cross-workgroup hazard).
// ---------------------------------------------------------------------------
__global__ void hbox_ab(float* __restrict__ aF, float* __restrict__ bF) {
  extern __shared__ float lds[];
  float* aL = lds;
  float* bL = lds + 16 * LDS_STRIDE;

  const int band = blockIdx.x;
  const int b    = blockIdx.y;
  const int r0   = band * 16;

  zero_guards(aL, bL);
  for (int idx = threadIdx.x; idx < 16 * WN; idx += 256) {
    const int rr = idx / WN;
    const int cc = idx - rr * WN;
    const size_t o = ((size_t)b * HN + r0 + rr) * WN + cc;
    aL[rr * LDS_STRIDE + GUARD + cc] = aF[o];
    bL[rr * LDS_STRIDE + GUARD + cc] = bF[o];
  }
  __syncthreads();

  const int lane  = threadIdx.x & 31;
  const int wave  = threadIdx.x >> 5;
  const int Mrow  = lane & 15;
  const int khalf = (lane >> 4) << 1;
  const int ncol  = lane & 15;
  const int rowOff = (lane >> 4) << 3;

  v2f bbA[KT_PER_TILE];
  make_band(bbA, khalf, ncol);
  const int arow = Mrow * LDS_STRIDE + GUARD;

  for (int t = 0; t < 10; ++t) {
    const int c0 = (wave * 10 + t) * 16;
    v8f accA = {0,0,0,0,0,0,0,0};
    v8f accB = {0,0,0,0,0,0,0,0};
    const int kbase = arow + c0 - RAD + khalf;
#pragma unroll
    for (int i = 0; i < KT_PER_TILE; ++i) {     // full unroll: bbA stays in VGPRs
      const int k = kbase + 4 * i;
      v2f fa = {aL[k], aL[k + 1]};
      v2f fb = {bL[k], bL[k + 1]};
      accA = wmma_acc(accA, fa, bbA[i]);
      accB = wmma_acc(accB, fb, bbA[i]);
    }
#pragma unroll
    for (int j = 0; j < 8; ++j) {
      const int row = r0 + j + rowOff;
      const size_t o = ((size_t)b * HN + row) * WN + (c0 + ncol);
      aF[o] = accA[j];
      bF[o] = accB[j];
    }
  }
}

// ---------------------------------------------------------------------------
// Top-k (k=983 per image): radix select on float bits (non-negative depths,
// so uint compare == float compare). Integer atomics -> deterministic.
// ---------------------------------------------------------------------------
__global__ void topk_init(unsigned* __restrict__ buf) {
  const int i = blockIdx.x * 256 + threadIdx.x;
  if (i < 4 * BN * 256 + BN * 4) buf[i] = 0u;
}

__global__ void topk_hist(const float* __restrict__ depth,
                          unsigned* __restrict__ hist4,
                          const unsigned* __restrict__ state, int round) {
  const int b = blockIdx.y;
  __shared__ unsigned lh[256];
  lh[threadIdx.x] = 0u;
  __syncthreads();
  const int s = 24 - 8 * round;
  const unsigned prefix = state[b * 4 + 0];
  const int stride = gridDim.x * 256;
  for (int i = blockIdx.x * 256 + threadIdx.x; i < HWN; i += stride) {
    const unsigned key = __float_as_uint(depth[(size_t)b * HWN + i]);
    const bool match = (round == 0) || ((key >> (s + 8)) == (prefix >> (s + 8)));
    if (match) atomicAdd(&lh[(key >> s) & 255u], 1u);
  }
  __syncthreads();
  const unsigned v = lh[threadIdx.x];
  if (v) atomicAdd(&hist4[((size_t)round * BN + b) * 256 + threadIdx.x], v);
}

__global__ void topk_select(const unsigned* __restrict__ hist4,
                            unsigned* __restrict__ state, int round) {
  if (threadIdx.x != 0) return;
  const int b = blockIdx.x;
  const int s = 24 - 8 * round;
  const unsigned prefix = state[b * 4 + 0];
  unsigned cum = state[b * 4 + 1];
  const unsigned* h = &hist4[((size_t)round * BN + b) * 256];
  for (int v = 255; v >= 0; --v) {
    const unsigned c = h[v];
    if (cum + c >= (unsigned)KTOP) {
      state[b * 4 + 0] = prefix | ((unsigned)v << s);
      state[b * 4 + 1] = cum;  // count strictly greater than new prefix
      return;
    }
    cum += c;
  }
}

// Deterministic per-block partial sums of colors for (key > thr) and
// (key == thr), plus the equal-count. Fixed-order in-block tree reduction.
__global__ void topk_partial(const float* __restrict__ depth,
                             const float* __restrict__ img,
                             const unsigned* __restrict__ state,
                             float* __restrict__ partial) {
  const int b   = blockIdx.y;
  const int blk = blockIdx.x;
  const unsigned thr = state[b * 4 + 0];
  const float* iR = img + (size_t)(b * 3 + 0) * HWN;
  const float* iG = img + (size_t)(b * 3 + 1) * HWN;
  const float* iB = img + (size_t)(b * 3 + 2) * HWN;
  float v0 = 0.f, v1 = 0.f, v2 = 0.f, v3 = 0.f, v4 = 0.f, v5 = 0.f, v6 = 0.f;
  const int base = blk * 4096;
#pragma unroll 4
  for (int j = 0; j < 16; ++j) {
    const int i = base + j * 256 + threadIdx.x;
    const unsigned key = __float_as_uint(depth[(size_t)b * HWN + i]);
    if (key >= thr) {
      const float R = iR[i], G = iG[i], Bc = iB[i];
      if (key > thr) { v0 += R; v1 += G; v2 += Bc; }
      else           { v3 += R; v4 += G; v5 += Bc; v6 += 1.f; }
    }
  }
  __shared__ float red[256 * 8];
  const int t = threadIdx.x;
  red[t * 8 + 0] = v0; red[t * 8 + 1] = v1; red[t * 8 + 2] = v2;
  red[t * 8 + 3] = v3; red[t * 8 + 4] = v4; red[t * 8 + 5] = v5;
  red[t * 8 + 6] = v6; red[t * 8 + 7] = 0.f;
  __syncthreads();
  for (int off = 128; off > 0; off >>= 1) {
    if (t < off) {
#pragma unroll
      for (int v = 0; v < 7; ++v) red[t * 8 + v] += red[(t + off) * 8 + v];
    }
    __syncthreads();
  }
  if (t == 0) {
#pragma unroll
    for (int v = 0; v < 7; ++v)
      partial[((size_t)b * PBLK + blk) * 8 + v] = red[v];
  }
}

__global__ void topk_finalize(const float* __restrict__ partial,
                              const unsigned* __restrict__ state,
                              float* __restrict__ Aatm) {
  if (threadIdx.x != 0) return;
  const int b = blockIdx.x;
  float sg0 = 0.f, sg1 = 0.f, sg2 = 0.f, se0 = 0.f, se1 = 0.f, se2 = 0.f, ce = 0.f;
  for (int blk = 0; blk < PBLK; ++blk) {   // fixed order -> deterministic
    const float* p = &partial[((size_t)b * PBLK + blk) * 8];
    sg0 += p[0]; sg1 += p[1]; sg2 += p[2];
    se0 += p[3]; se1 += p[4]; se2 += p[5]; ce += p[6];
  }
  const int n_gt = (int)state[b * 4 + 1];
  const float m = (float)(KTOP - n_gt);          // #ties to include (>=1)
  const float w = (ce > 0.f) ? (m / ce) : 0.f;   // fractional tie weighting
  Aatm[b * 4 + 0] = (sg0 + w * se0) / (float)KTOP;
  Aatm[b * 4 + 1] = (sg1 + w * se1) / (float)KTOP;
  Aatm[b * 4 + 2] = (sg2 + w * se2) / (float)KTOP;
}

// ---------------------------------------------------------------------------
// Kernel D: vertical box of hbox(a), hbox(b) -> mean_a, mean_b;
// t = mean_a*gray + mean_b; out = img*t + A*(1-t).
// ---------------------------------------------------------------------------
__global__ void vbox_final(const float* __restrict__ aF, const float* __restrict__ bF,
                           const float* __restrict__ img,
                           const float* __restrict__ Aatm,
                           float* __restrict__ out) {
  const int gid = blockIdx.x * blockDim.x + threadIdx.x;
  if (gid >= BN * WN * NCHUNK) return;
  const int x     = gid % WN;
  const int rest  = gid / WN;
  const int b     = rest % BN;
  const int chunk = rest / BN;
  const int y0    = chunk * CHUNK;
  const size_t base = (size_t)b * HN * WN + x;
  const float A0 = Aatm[b * 4 + 0];
  const float A1 = Aatm[b * 4 + 1];
  const float A2 = Aatm[b * 4 + 2];

  float sA = 0.f, sB = 0.f;
  for (int yy = y0 - RAD; yy < y0 + RAD; ++yy) {
    if (yy >= 0 && yy < HN) {
      const size_t o = base + (size_t)yy * WN;
      sA += aF[o]; sB += bF[o];
    }
  }
  for (int y = y0; y < y0 + CHUNK; ++y) {
    const int ya = y + RAD;
    if (ya < HN) {
      const size_t o = base + (size_t)ya * WN;
      sA += aF[o]; sB += bF[o];
      if (ya + 16 < HN)
        __builtin_prefetch(aF + base + (size_t)(ya + 16) * WN, 0, 1);
    }
    const float ma = sA * INV_N;
    const float mb = sB * INV_N;
    const size_t ib = ((size_t)(b * 3) * HN + y) * WN + x;
    const float R  = img[ib];
    const float G  = img[ib + (size_t)HN * WN];
    const float Bc = img[ib + 2 * (size_t)HN * WN];
    const float g = 0.299f * R + 0.587f * G + 0.114f * Bc;
    const float t = ma * g + mb;
    const float omt = 1.0f - t;
    out[ib]                       = R * t + A0 * omt;
    out[ib + (size_t)HN * WN]     = G * t + A1 * omt;
    out[ib + 2 * (size_t)HN * WN] = Bc * t + A2 * omt;
    const int ys = y - RAD;
    if (ys >= 0) {
      const size_t o = base + (size_t)ys * WN;
      sA -= aF[o]; sB -= bF[o];
    }
  }
}

// ---------------------------------------------------------------------------
extern "C" void kernel_launch(void* const* d_in, const int* in_sizes, int n_in,
                              void* d_out, int out_size, void* d_ws, size_t ws_size,
                              hipStream_t stream) {
  const float* img   = (const float*)d_in[0];   // (8,3,768,1280) f32
  const float* depth = (const float*)d_in[1];   // (8,1,768,1280) f32
  float* out = (float*)d_out;                   // (8,3,768,1280) f32

  float* ws = (float*)d_ws;
  const size_t F = (size_t)BN * HN * WN;        // 7,864,320 floats
  float* fI  = ws + 0 * F;
  float* fP  = ws + 1 * F;
  float* fIP = ws + 2 * F;
  float* fII = ws + 3 * F;
  float* aF  = ws + 4 * F;
  float* bF  = ws + 5 * F;
  unsigned* hist4 = (unsigned*)(ws + 6 * F);          // [4][B][256]
  unsigned* state = hist4 + 4 * BN * 256;             // [B][4]
  float* partial  = (float*)(state + BN * 4);         // [B][PBLK][8]
  float* Aatm     = partial + (size_t)BN * PBLK * 8;  // [B][4]

  const size_t ldsBytes = (size_t)2 * 16 * LDS_STRIDE * sizeof(float); // ~174KB

  // Guided filter, phase 1: fields + horizontal box (WMMA), vertical box + a,b.
  prep_hbox<<<dim3(HN / 16, BN), 256, ldsBytes, stream>>>(img, depth, fI, fP, fIP, fII);
  vbox_ab<<<(BN * WN * NCHUNK) / 256, 256, 0, stream>>>(fI, fP, fIP, fII, aF, bF);
  // Phase 2: box of a,b (horizontal via WMMA in place; vertical fused below).
  hbox_ab<<<dim3(HN / 16, BN), 256, ldsBytes, stream>>>(aF, bF);

  // Atmospheric light: exact radix-select top-k, deterministic reductions.
  topk_init<<<(4 * BN * 256 + BN * 4 + 255) / 256, 256, 0, stream>>>(hist4);
  for (int r = 0; r < 4; ++r) {
    topk_hist<<<dim3(120, BN), 256, 0, stream>>>(depth, hist4, state, r);
    topk_select<<<BN, 32, 0, stream>>>(hist4, state, r);
  }
  topk_partial<<<dim3(PBLK, BN), 256, 0, stream>>>(depth, img, state, partial);
  topk_finalize<<<BN, 32, 0, stream>>>(partial, state, Aatm);

  // Vertical box of a,b + transmission + composite.
  vbox_final<<<(BN * WN * NCHUNK) / 256, 256, 0, stream>>>(aF, bF, img, Aatm, out);
}